// BackProjection_40621800685677
// MI455X (gfx1250) — compile-verified
//
#include <hip/hip_runtime.h>

// ---------------------------------------------------------------------------
// Siddon ray-driven backprojection for MI455X (gfx1250, wave32).
//
// This workload is a serial-per-ray scatter: no matmul structure, so the
// optimal CDNA5 path is wave32 lanes (1 lane = 1 LOR), IEEE f32 geometry
// matching the reference, and no-return GLOBAL_ATOMIC_ADD_F32 into the
// L2-resident 8MB image (192MB L2 => atomics never touch HBM).
// ---------------------------------------------------------------------------

#define GX 128
#define GY 128
#define GZ 128
#define N_STEPS (GX + GY + GZ + 4)   // 388, matches reference scan length

__global__ __launch_bounds__(256) void bp_zero_kernel(float4* __restrict__ out, int n4) {
    int i = blockIdx.x * blockDim.x + threadIdx.x;
    if (i < n4) out[i] = make_float4(0.f, 0.f, 0.f, 0.f);   // global_store_b128
}

__global__ __launch_bounds__(256) void siddon_bp_kernel(const float* __restrict__ lors,
                                                        const float* __restrict__ proj,
                                                        float* __restrict__ out,
                                                        int n) {
    const int i = blockIdx.x * blockDim.x + threadIdx.x;
    if (i >= n) return;

    // gfx1250 prefetch path (global_prefetch_b8): warm the LOR + proj lines.
    __builtin_prefetch(lors + (size_t)i * 6, 0, 0);
    __builtin_prefetch(proj + i, 0, 0);

    const float corner = -200.0f;   // ORIGIN - 0.5*SIZE  (exact in f32)
    const float hi     =  200.0f;
    const float vsize  =  3.125f;   // 400/128            (exact in f32)
    const float dimf[3] = { (float)GX, (float)GY, (float)GZ };
    const float INF = __builtin_inff();

    // LOR endpoints: stride 6 floats = 24B, 8B-aligned => three b64 loads.
    const float2* l2 = (const float2*)lors;
    float2 v0 = l2[(size_t)i * 3 + 0];
    float2 v1 = l2[(size_t)i * 3 + 1];
    float2 v2 = l2[(size_t)i * 3 + 2];

    float p1[3] = { v0.x, v0.y, v1.x };
    float p2[3] = { v1.y, v2.x, v2.y };

    float d[3], invd[3];
    bool  zero[3];
    float aminv = 0.0f;   // reference: max(max_ax, 0)
    float amaxv = 1.0f;   // reference: min(min_ax, 1)
    bool  miss  = false;

    #pragma unroll
    for (int ax = 0; ax < 3; ++ax) {
        d[ax]    = p2[ax] - p1[ax];
        zero[ax] = (d[ax] == 0.0f);
        invd[ax] = zero[ax] ? 0.0f : (1.0f / d[ax]);          // IEEE f32 division
        if (zero[ax]) {
            miss = miss || (p1[ax] <= corner) || (p1[ax] >= hi);
            // a_min_ax = -inf, a_max_ax = +inf: no-ops on the folds below.
        } else {
            float a1 = (corner - p1[ax]) * invd[ax];
            float a2 = (hi     - p1[ax]) * invd[ax];
            aminv = fmaxf(aminv, fminf(a1, a2));
            amaxv = fminf(amaxv, fmaxf(a1, a2));
        }
    }

    const bool valid = (amaxv > aminv) && !miss;
    if (!valid) return;   // reference scan contributes exactly nothing

    const float ray_len = sqrtf(d[0]*d[0] + d[1]*d[1] + d[2]*d[2]);
    const float pw      = proj[i];

    float alpha = aminv;

    for (int s = 0; s < N_STEPS; ++s) {
        // next axis-plane crossing (reference: epsilon'd floor/ceil stepping)
        float m = INF;
        #pragma unroll
        for (int ax = 0; ax < 3; ++ax) {
            float x   = p1[ax] + alpha * d[ax];
            float t   = (x - corner) / vsize;                 // true division
            float nxt = (d[ax] > 0.0f) ? (floorf(t + 1e-5f) + 1.0f)
                                       : (ceilf (t - 1e-5f) - 1.0f);
            float a   = (corner + nxt * vsize - p1[ax]) * invd[ax];
            a = zero[ax] ? INF : a;
            m = fminf(m, a);
        }
        // a_next = clip(m, alpha, amax)
        float a_next = fminf(fmaxf(m, alpha), amaxv);
        float a_mid  = 0.5f * (alpha + a_next);

        // midpoint voxel; bounds-check in float domain before int cast
        bool inside = true;
        int  idx    = 0;
        #pragma unroll
        for (int ax = 0; ax < 3; ++ax) {
            float vf = floorf((p1[ax] + a_mid * d[ax] - corner) / vsize);
            inside = inside && (vf >= 0.0f) && (vf < dimf[ax]);
            idx = idx * 128 + (int)vf;
        }

        if (inside && (a_next > alpha)) {
            // reference multiply order: w = (a_next-alpha)*ray_len; contrib = w*proj
            float contrib = ((a_next - alpha) * ray_len) * pw;
            atomicAdd(out + idx, contrib);   // no-return global_atomic_add_f32
        }

        // Past this point the reference scan can only produce w == 0:
        // a_next == alpha repeats forever; a_next == amax pins there.
        if (!(a_next > alpha) || (a_next >= amaxv)) break;
        alpha = a_next;
    }
}

extern "C" void kernel_launch(void* const* d_in, const int* in_sizes, int n_in,
                              void* d_out, int out_size, void* d_ws, size_t ws_size,
                              hipStream_t stream) {
    (void)d_ws; (void)ws_size; (void)n_in;
    // setup_inputs order: image (unused), lors (N,6), proj (N,)
    const float* lors = (const float*)d_in[1];
    const float* proj = (const float*)d_in[2];
    float* out = (float*)d_out;
    const int n_lors = in_sizes[2];          // 100000

    // 1) zero the 128^3 image (harness poisons d_out; reference starts from 0)
    int n4 = out_size / 4;                   // out_size = 2^21, divisible by 4
    bp_zero_kernel<<<(n4 + 255) / 256, 256, 0, stream>>>((float4*)out, n4);

    // 2) trace all LORs, one wave32 lane per LOR (3125 waves -> fills device)
    siddon_bp_kernel<<<(n_lors + 255) / 256, 256, 0, stream>>>(lors, proj, out, n_lors);
}